// Generator_57140244906364
// MI455X (gfx1250) — compile-verified
//
#include <hip/hip_runtime.h>
#include <cstddef>

// ------------------------------------------------------------------
// Tsit5 integrator for B=256 x DOF=64 second-order systems on gfx1250.
// Batched 64x64 matvecs done with V_WMMA_F32_16X16X4_F32 (full f32).
// One wave (32 threads) per block owns 16 batch rows; matrices, state
// and RK acceleration slopes live in LDS.
// kt-outer / nt-inner: A-fragments built once per kt and shared by the
// 4 independent accumulator chains (latency hiding + 4x less prep).
// ------------------------------------------------------------------

typedef __attribute__((ext_vector_type(2))) float v2f;
typedef __attribute__((ext_vector_type(8))) float v8f;

#define DOF    64
#define ROWS   16
#define STRIDE 68     // padded row stride (floats): bank-conflict-free
#define NSTAGE 6

template<int N> struct IC { static constexpr int value = N; };

// Tsit5 tableau + composite coefficients:
//   v_s = v + h * sum_j A[s][j] * ka_j
//   d_s = d + h*C[s]*v + h^2 * sum_j Q[s][j] * ka_j   (C[s]=sum A, Q = A*A)
//   v'  = v + h * sum B_i ka_i
//   d'  = d + h*SB*v + h^2 * sum QB_l ka_l
struct Coef {
  float A[7][7]; float C[7]; float Q[7][7]; float B[7]; float QB[7]; float SB;
  constexpr Coef() : A{}, C{}, Q{}, B{}, QB{}, SB(0.f) {
    double a[7][7] = {};
    a[2][1] = 0.161;
    a[3][1] = -0.008480655492356989; a[3][2] = 0.335480655492357;
    a[4][1] = 2.8971530571054935;    a[4][2] = -6.359448489975075;  a[4][3] = 4.3622954328695815;
    a[5][1] = 5.325864828439257;     a[5][2] = -11.748883564062828; a[5][3] = 7.4955393428898365;  a[5][4] = -0.09249506636175525;
    a[6][1] = 5.86145544294642;      a[6][2] = -12.92096931784711;  a[6][3] = 8.159367898576159;   a[6][4] = -0.071584973281401; a[6][5] = -0.028269050394068383;
    double b[7] = {0.0, 0.09646076681806523, 0.01, 0.4798896504144996,
                   1.379008574103742, -3.290069515436081, 2.324710524099774};
    for (int s = 2; s <= 6; s++) {
      double c = 0.0;
      for (int j = 1; j < s; j++) { c += a[s][j]; A[s][j] = (float)a[s][j]; }
      C[s] = (float)c;
      for (int l = 1; l < s; l++) {
        double q = 0.0;
        for (int j = l + 1; j < s; j++) q += a[s][j] * a[j][l];
        Q[s][l] = (float)q;
      }
    }
    double sb = 0.0;
    for (int i = 1; i <= 6; i++) { B[i] = (float)b[i]; sb += b[i]; }
    SB = (float)sb;
    for (int l = 1; l <= 6; l++) {
      double q = 0.0;
      for (int i = l + 1; i <= 6; i++) q += b[i] * a[i][l];
      QB[l] = (float)q;
    }
  }
};

__global__ __launch_bounds__(32)
void tsit5_wmma_kernel(const float* __restrict__ cprop, const float* __restrict__ knl,
                       const float* __restrict__ Minv,  const float* __restrict__ Mt,
                       const float* __restrict__ Kt,    const float* __restrict__ Ct,
                       const float* __restrict__ ft,    const float* __restrict__ tspan,
                       const float* __restrict__ d0,    const float* __restrict__ v0,
                       float* __restrict__ out, int T)
{
  constexpr Coef CF{};

  // B-operand fragments for M0 = -Kt^T and M1 = -Ceff^T:
  //  fragment (m,kt,nt): 4x16 tile; lanes0-15 -> N, K={4kt,4kt+1};
  //  lanes16-31 -> N, K={4kt+2,4kt+3}; VGPR j selects the pair element.
  __shared__ float Bfrag[2 * 16 * 4 * 64];              // 32 KB
  __shared__ float dL[ROWS * STRIDE];                   // base displacement
  __shared__ float vL[ROWS * STRIDE];                   // base velocity
  __shared__ float kaL[NSTAGE][ROWS * STRIDE];          // acceleration slopes

  const int lane = threadIdx.x;
  const int lo = lane & 15;
  const int hi = lane >> 4;
  const int r0 = blockIdx.x * ROWS;
  const int TD = T * DOF;

  const float cp0 = cprop[0], cp1 = cprop[1];
  const float kn0 = knl[0], kn1 = knl[1], kn2 = knl[2], kn3 = knl[3], kn4 = knl[4];

  // ---- one-time: pack negated transposed matrices into B-fragment layout ----
  for (int i = lane; i < 2 * 16 * 4 * 64; i += 32) {
    const int j  = i & 1;
    const int l  = (i >> 1) & 31;
    const int nt = (i >> 6) & 3;
    const int kt = (i >> 8) & 15;
    const int m  = i >> 12;
    const int n  = nt * 16 + (l & 15);
    const int k  = kt * 4 + ((l >> 4) << 1) + j;
    const float kv = Kt[n * DOF + k];
    float val = -kv;                                    // -Kt^T element (K=k,N=n)
    if (m) val = -(Ct[n * DOF + k] + cp0 * Mt[n * DOF + k] + cp1 * kv); // -Ceff^T
    Bfrag[i] = val;
  }
  // ---- initial state ----
  for (int i = lane; i < ROWS * DOF; i += 32) {
    const int r = i >> 6, k = i & (DOF - 1);
    dL[r * STRIDE + k] = d0[(r0 + r) * DOF + k];
    vL[r * STRIDE + k] = v0[(r0 + r) * DOF + k];
  }
  float ftv[4], miv[4];
  #pragma unroll
  for (int nt = 0; nt < 4; nt++) {
    const int n = nt * 16 + lo;
    ftv[nt] = ft[n];
    miv[nt] = Minv[n * DOF + n];
  }
  __syncthreads();

  // ---- one RK stage: A-fragments built once per kt, 4 independent acc chains ----
  auto stage = [&](auto Sc, float h, float h2, float* op) {
    constexpr int S = decltype(Sc)::value;
    v8f acc[4];
    #pragma unroll
    for (int nt = 0; nt < 4; nt++) {
      #pragma unroll
      for (int j = 0; j < 8; j++) acc[nt][j] = ftv[nt]; // F starts at ft[n]
    }
    #pragma unroll
    for (int kt = 0; kt < 16; kt++) {
      const int base = lo * STRIDE + kt * 4 + hi * 2;   // A-frag element (row=lo, k)
      const v2f dd = *(const v2f*)(dL + base);
      const v2f vv = *(const v2f*)(vL + base);
      v2f ad, av;
      if constexpr (S == 1) { ad = dd; av = vv; }
      else {
        v2f sA = {0.f, 0.f}, sQ = {0.f, 0.f};
        #pragma unroll
        for (int j = 1; j < S; j++) {
          const v2f kj = *(const v2f*)(&kaL[j - 1][base]);
          sA += CF.A[S][j] * kj;
          sQ += CF.Q[S][j] * kj;
        }
        av = vv + h * sA;
        ad = dd + (h * CF.C[S]) * vv + h2 * sQ;
      }
      v2f bd[4], bv[4];
      #pragma unroll
      for (int nt = 0; nt < 4; nt++) {
        bd[nt] = *(const v2f*)(Bfrag + (kt * 4 + nt) * 64 + lane * 2);
        bv[nt] = *(const v2f*)(Bfrag + ((16 + kt) * 4 + nt) * 64 + lane * 2);
      }
      // d-contributions first, then v-contributions: each accumulator's
      // consecutive updates are separated by 3 independent WMMAs.
      #pragma unroll
      for (int nt = 0; nt < 4; nt++)
        acc[nt] = __builtin_amdgcn_wmma_f32_16x16x4_f32(false, ad, false, bd[nt],
                                                        (short)0, acc[nt], false, false);
      #pragma unroll
      for (int nt = 0; nt < 4; nt++)
        acc[nt] = __builtin_amdgcn_wmma_f32_16x16x4_f32(false, av, false, bv[nt],
                                                        (short)0, acc[nt], false, false);
    }
    // nonlinear spring acts only on column 63 (lanes 15/31 of C layout, nt==3)
    #pragma unroll
    for (int j = 0; j < 8; j++) {
      const int idx = (j + hi * 8) * STRIDE + (DOF - 1);
      float dl = dL[idx];
      if constexpr (S != 1) {
        float sQ = 0.f;
        #pragma unroll
        for (int jj = 1; jj < S; jj++) sQ += CF.Q[S][jj] * kaL[jj - 1][idx];
        dl += h * CF.C[S] * vL[idx] + h2 * sQ;
      }
      const float f = dl * (kn0 + dl * (kn1 + dl * (kn2 + dl * (kn3 + dl * kn4))));
      acc[3][j] -= (lo == 15) ? f : 0.f;
    }
    // scale by Minv diag, store slope (and optionally the accel output)
    #pragma unroll
    for (int nt = 0; nt < 4; nt++) {
      #pragma unroll
      for (int j = 0; j < 8; j++) {
        const float kv = acc[nt][j] * miv[nt];          // accel = F * Minv_diag
        const int row = j + hi * 8;
        kaL[S - 1][row * STRIDE + nt * 16 + lo] = kv;
        if (op) op[row * TD + nt * 16 + lo] = kv;       // stage-1 accel == output
      }
    }
  };

  // ---- combine stages: y <- y + h*sum(b_i k_i), using composite coefficients ----
  auto finish = [&](float h, float h2) {
    for (int i = lane; i < ROWS * (DOF / 2); i += 32) {
      const int base = (i >> 5) * STRIDE + (i & 31) * 2;
      const v2f dd = *(const v2f*)(dL + base);
      const v2f vv = *(const v2f*)(vL + base);
      v2f sB = {0.f, 0.f}, sQB = {0.f, 0.f};
      #pragma unroll
      for (int j = 1; j <= NSTAGE; j++) {
        const v2f kj = *(const v2f*)(&kaL[j - 1][base]);
        sB  += CF.B[j]  * kj;
        sQB += CF.QB[j] * kj;
      }
      *(v2f*)(vL + base) = vv + h * sB;
      *(v2f*)(dL + base) = dd + (h * CF.SB) * vv + h2 * sQB;
    }
  };

  float* outBase = out + (size_t)r0 * TD;
  for (int t = 0; t < T - 1; t++) {
    const float h  = 0.5f * (tspan[t + 1] - tspan[t]);  // dt / SUBSTEPS
    const float h2 = h * h;
    // substep 0: stage-1 slope is the acceleration at stored time t
    stage(IC<1>{}, h, h2, outBase + t * DOF);
    stage(IC<2>{}, h, h2, nullptr);
    stage(IC<3>{}, h, h2, nullptr);
    stage(IC<4>{}, h, h2, nullptr);
    stage(IC<5>{}, h, h2, nullptr);
    stage(IC<6>{}, h, h2, nullptr);
    finish(h, h2);
    // substep 1
    stage(IC<1>{}, h, h2, nullptr);
    stage(IC<2>{}, h, h2, nullptr);
    stage(IC<3>{}, h, h2, nullptr);
    stage(IC<4>{}, h, h2, nullptr);
    stage(IC<5>{}, h, h2, nullptr);
    stage(IC<6>{}, h, h2, nullptr);
    finish(h, h2);
  }
  // acceleration at the final stored time point
  stage(IC<1>{}, 0.f, 0.f, outBase + (size_t)(T - 1) * DOF);
}

extern "C" void kernel_launch(void* const* d_in, const int* in_sizes, int n_in,
                              void* d_out, int out_size, void* d_ws, size_t ws_size,
                              hipStream_t stream) {
  (void)n_in; (void)out_size; (void)d_ws; (void)ws_size;
  const float* cprop = (const float*)d_in[0];
  const float* knl   = (const float*)d_in[1];
  const float* Minv  = (const float*)d_in[2];
  const float* Mt    = (const float*)d_in[3];
  const float* Kt    = (const float*)d_in[4];
  const float* Ct    = (const float*)d_in[5];
  const float* ft    = (const float*)d_in[6];
  const float* tspan = (const float*)d_in[7];
  const float* d0    = (const float*)d_in[8];
  const float* v0    = (const float*)d_in[9];
  float* out = (float*)d_out;

  const int T = in_sizes[7];            // 1000
  const int B = in_sizes[8] / DOF;      // 256

  dim3 grid(B / ROWS), block(32);       // 16 blocks x 1 wave
  tsit5_wmma_kernel<<<grid, block, 0, stream>>>(cprop, knl, Minv, Mt, Kt, Ct, ft,
                                                tspan, d0, v0, out, T);
}